// StatePredictor_52140902974173
// MI455X (gfx1250) — compile-verified
//
#include <hip/hip_runtime.h>

// Problem constants (match reference)
#define NB   2
#define NPT  8192
#define KNN  11          // k+1 (self included)
#define LEN  66          // 3*(K+1) + 3*K + I = 33 + 30 + 3
#define LENP 68          // padded to multiple of 4 (68%64==4 -> conflict-free LDS reads)
#define QPB  64          // queries per block (64 threads = 2 wave32)
#define CH   1024        // candidate chunk staged in LDS (1024*3*4 = 12KB)
#define AIPW 12          // async b128 issues per thread per chunk: (CH*3/4)/QPB

typedef __attribute__((ext_vector_type(2))) float v2f;
typedef __attribute__((ext_vector_type(8))) float v8f;

// Builtin-mandated types (from hipcc diagnostic): arg0 is v4i in AS(1), arg1 LDS side.
typedef int v4i __attribute__((vector_size(16)));
typedef __attribute__((address_space(1))) v4i* gp_t;
typedef __attribute__((address_space(3))) v4i* lp_t;

// Stream one 12KB candidate chunk global -> LDS via the CDNA5 async path.
// Each thread issues AIPW=12 b128 transfers => 12 ASYNCcnt increments per wave.
__device__ __forceinline__ void async_copy_chunk(const float* gsrc, float* ldst, int tid) {
  gp_t g = (gp_t)gsrc;
  lp_t l = (lp_t)ldst;
#pragma unroll
  for (int i = 0; i < AIPW; ++i) {
    const int e = tid + i * QPB;
    __builtin_amdgcn_global_load_async_to_lds_b128(g + e, l + e, 0, 0);
  }
}

__global__ __launch_bounds__(QPB)
void StatePredictor_52140902974173_kernel(
    const float* __restrict__ pos, const float* __restrict__ vel,
    const float* __restrict__ cfg, const float* __restrict__ W1,
    const float* __restrict__ b1,  const float* __restrict__ W2,
    const float* __restrict__ b2,  const float* __restrict__ W3,
    const float* __restrict__ b3,  float* __restrict__ out)
{
  // fpool: double-buffered candidate staging (2 x CH*3 = 6144 floats, 24KB),
  // later reused as the per-row feature matrix (QPB x LENP = 4352 floats).
  __shared__ __attribute__((aligned(16))) float fpool[2 * CH * 3];
  __shared__ float h1s[QPB / 32][16 * 33];   // 16x32 tile, padded stride 33
  __shared__ float h2s[QPB / 32][16 * 17];   // 16x16 tile, padded stride 17

  const int tid   = threadIdx.x;
  const int qbase = blockIdx.x * QPB;        // block never straddles a batch (QPB | NPT)
  const int b     = qbase / NPT;
  const int g     = qbase + tid;             // flat row index b*N + n
  const float* posb = pos + (size_t)b * NPT * 3;
  const float* velb = vel + (size_t)b * NPT * 3;

  const float qx = pos[(size_t)g * 3 + 0];
  const float qy = pos[(size_t)g * 3 + 1];
  const float qz = pos[(size_t)g * 3 + 2];

  // ---------------- Phase 1: exact kNN, async double-buffered streaming -----
  float bd[KNN]; int bi[KNN];
#pragma unroll
  for (int j = 0; j < KNN; ++j) { bd[j] = 3.4e38f; bi[j] = 0; }

  auto consider = [&](float cx, float cy, float cz, int idx) {
    const float dx = cx - qx, dy = cy - qy, dz = cz - qz;
    const float d  = fmaf(dx, dx, fmaf(dy, dy, dz * dz));
    if (d < bd[KNN - 1]) {                             // stable insert (strict <)
      bd[KNN - 1] = d; bi[KNN - 1] = idx;
#pragma unroll
      for (int s = KNN - 1; s > 0; --s) {
        const float pd = bd[s - 1]; const int pi = bi[s - 1];
        if (bd[s] < pd) {
          bd[s - 1] = bd[s]; bi[s - 1] = bi[s];
          bd[s] = pd;        bi[s] = pi;
        }
      }
    }
  };

  async_copy_chunk(posb, fpool, tid);                  // prime buffer 0
  int buf = 0;
  for (int c0 = 0; c0 < NPT; c0 += CH) {
    if (c0 + CH < NPT) {                               // overlap: stream next chunk
      async_copy_chunk(posb + (size_t)(c0 + CH) * 3, fpool + (buf ^ 1) * (CH * 3), tid);
      __builtin_amdgcn_s_wait_asynccnt(AIPW);          // current chunk has landed
    } else {
      __builtin_amdgcn_s_wait_asynccnt(0);
    }
    __syncthreads();                                   // both waves' halves visible

    const float4* cp = (const float4*)(fpool + buf * (CH * 3));
    for (int j4 = 0; j4 < CH / 4; ++j4) {              // 3 x ds_load_b128 -> 4 candidates
      const float4 p0 = cp[3 * j4 + 0];
      const float4 p1 = cp[3 * j4 + 1];
      const float4 p2 = cp[3 * j4 + 2];
      const int ci = c0 + 4 * j4;
      consider(p0.x, p0.y, p0.z, ci + 0);
      consider(p0.w, p1.x, p1.y, ci + 1);
      consider(p1.z, p1.w, p2.x, ci + 2);
      consider(p2.y, p2.z, p2.w, ci + 3);
    }
    __syncthreads();                                   // scan done before overwrite
    buf ^= 1;
  }

  // ---------------- Phase 2: build 66-wide feature rows in LDS --------------
  float* fr = &fpool[tid * LENP];
#pragma unroll
  for (int j = 0; j < KNN; ++j) {                      // neighbor velocities (incl self)
    const size_t nb = (size_t)bi[j] * 3;
    fr[3 * j + 0] = velb[nb + 0];
    fr[3 * j + 1] = velb[nb + 1];
    fr[3 * j + 2] = velb[nb + 2];
  }
#pragma unroll
  for (int j = 1; j < KNN; ++j) {                      // position offsets (excl self)
    const size_t nb = (size_t)bi[j] * 3;
    fr[33 + 3 * (j - 1) + 0] = posb[nb + 0] - qx;
    fr[33 + 3 * (j - 1) + 1] = posb[nb + 1] - qy;
    fr[33 + 3 * (j - 1) + 2] = posb[nb + 2] - qz;
  }
  fr[63] = cfg[b * 3 + 0]; fr[64] = cfg[b * 3 + 1]; fr[65] = cfg[b * 3 + 2];
  fr[66] = 0.f; fr[67] = 0.f;                          // K-pad for 4-wide WMMA steps
  __syncthreads();

  // ---------------- Phase 3: MLP via V_WMMA_F32_16X16X4_F32 -----------------
  // A 16x4 layout: lane (lm,hi) holds row lm, K = 4*kk + 2*hi + {0,1}
  // B  4x16 layout: lane (lm,hi) holds col lm, K = 4*kk + 2*hi + {0,1}
  // C/D 16x16:      lane (lm,hi) holds col lm, rows r + 8*hi (r = VGPR 0..7)
  const int w    = tid >> 5;                           // wave id (wave32)
  const int lane = tid & 31;
  const int lm   = lane & 15;
  const int hi   = lane >> 4;

  for (int t = 0; t < 2; ++t) {                        // two 16-row tiles per wave
    const int rb = w * 32 + t * 16;                    // block-local row base

    // Layer 1: [16 x 66] @ [66 x 32], two 16-col tiles, 17 k-steps
    for (int ct = 0; ct < 2; ++ct) {
      v8f acc = {};
      for (int kk = 0; kk < LENP / 4; ++kk) {
        const int ka = kk * 4 + 2 * hi;
        // Clamped, unconditional loads (no exec-predicated branches in the loop);
        // K-pad rows select to zero via v_cndmask.
        const int k0 = (ka     < LEN) ? ka     : LEN - 1;
        const int k1 = (ka + 1 < LEN) ? ka + 1 : LEN - 1;
        v2f a, bf;
        a.x  = fpool[(rb + lm) * LENP + ka];
        a.y  = fpool[(rb + lm) * LENP + ka + 1];
        const float w0  = W1[k0 * 32 + ct * 16 + lm];
        const float w1v = W1[k1 * 32 + ct * 16 + lm];
        bf.x = (ka     < LEN) ? w0  : 0.f;
        bf.y = (ka + 1 < LEN) ? w1v : 0.f;
        acc = __builtin_amdgcn_wmma_f32_16x16x4_f32(false, a, false, bf,
                                                    (short)0, acc, false, false);
      }
      const float bb = b1[ct * 16 + lm];
#pragma unroll
      for (int r = 0; r < 8; ++r)                      // D-layout -> LDS (transpose buf)
        h1s[w][(r + 8 * hi) * 33 + ct * 16 + lm] = acc[r] + bb;
    }
    __syncthreads();

    // Layer 2: [16 x 32] @ [32 x 16], 8 k-steps
    {
      v8f acc = {};
      for (int kk = 0; kk < 8; ++kk) {
        const int ka = kk * 4 + 2 * hi;
        v2f a, bf;
        a.x  = h1s[w][lm * 33 + ka];
        a.y  = h1s[w][lm * 33 + ka + 1];
        bf.x = W2[(ka    ) * 16 + lm];
        bf.y = W2[(ka + 1) * 16 + lm];
        acc = __builtin_amdgcn_wmma_f32_16x16x4_f32(false, a, false, bf,
                                                    (short)0, acc, false, false);
      }
      const float bb = b2[lm];
#pragma unroll
      for (int r = 0; r < 8; ++r)
        h2s[w][(r + 8 * hi) * 17 + lm] = acc[r] + bb;
    }
    __syncthreads();

    // Layer 3: [16 x 16] @ [16 x 6 (padded to 16)], 4 k-steps
    {
      const int lmc = (lm < 6) ? lm : 5;               // clamp col, select to 0
      v8f acc = {};
      for (int kk = 0; kk < 4; ++kk) {
        const int ka = kk * 4 + 2 * hi;
        v2f a, bf;
        a.x  = h2s[w][lm * 17 + ka];
        a.y  = h2s[w][lm * 17 + ka + 1];
        const float w0  = W3[(ka    ) * 6 + lmc];
        const float w1v = W3[(ka + 1) * 6 + lmc];
        bf.x = (lm < 6) ? w0  : 0.f;
        bf.y = (lm < 6) ? w1v : 0.f;
        acc = __builtin_amdgcn_wmma_f32_16x16x4_f32(false, a, false, bf,
                                                    (short)0, acc, false, false);
      }
      if (lm < 6) {                                    // divergence only around stores
        const float bb = b3[lm];
#pragma unroll
        for (int r = 0; r < 8; ++r) {
          const int gr = qbase + rb + r + 8 * hi;
          float v = acc[r] + bb;
          if (lm < 3) v += pos[(size_t)gr * 3 + lm];   // pred[:, :3] += position
          out[(size_t)gr * 6 + lm] = v;
        }
      }
    }
    __syncthreads();
  }
}

extern "C" void kernel_launch(void* const* d_in, const int* in_sizes, int n_in,
                              void* d_out, int out_size, void* d_ws, size_t ws_size,
                              hipStream_t stream) {
  (void)in_sizes; (void)n_in; (void)out_size; (void)d_ws; (void)ws_size;
  const float* pos = (const float*)d_in[0];
  const float* vel = (const float*)d_in[1];
  const float* cfg = (const float*)d_in[2];
  const float* W1  = (const float*)d_in[3];
  const float* b1  = (const float*)d_in[4];
  const float* W2  = (const float*)d_in[5];
  const float* b2  = (const float*)d_in[6];
  const float* W3  = (const float*)d_in[7];
  const float* b3  = (const float*)d_in[8];
  float* out = (float*)d_out;

  const int blocks = (NB * NPT) / QPB;   // 256 blocks x 64 threads
  StatePredictor_52140902974173_kernel<<<dim3(blocks), dim3(QPB), 0, stream>>>(
      pos, vel, cfg, W1, b1, W2, b2, W3, b3, out);
}